// Symbolic_KANLayer_90555090469173
// MI455X (gfx1250) — compile-verified
//
#include <hip/hip_runtime.h>

typedef __attribute__((ext_vector_type(2))) float v2f;
typedef __attribute__((ext_vector_type(4))) float f4;
typedef __attribute__((ext_vector_type(8))) float v8f;

#define NB   2048
#define NIN  256
#define NOUT 256

// ---------------------------------------------------------------------------
// Kernel 1: fold affine+mask into W = m*a2*a0 and Bias = m*(a2*a1+a3)
// ---------------------------------------------------------------------------
__global__ void prep_kernel(const float* __restrict__ affine,
                            const float* __restrict__ mask,
                            float* __restrict__ W,
                            float* __restrict__ Bias) {
    int idx = blockIdx.x * blockDim.x + threadIdx.x;   // o*NIN + i, 0..65535
    f4 a = *(const f4*)(affine + (long long)idx * 4);  // a0,a1,a2,a3
    float m = mask[idx];
    W[idx]    = m * a.z * a.x;
    Bias[idx] = m * fmaf(a.z, a.y, a.w);
}

// ---------------------------------------------------------------------------
// Kernel 2: bsum[o] = sum_i Bias[o,i]   (one block per o, LDS reduction)
// ---------------------------------------------------------------------------
__global__ void biassum_kernel(const float* __restrict__ Bias,
                               float* __restrict__ bsum) {
    __shared__ float s[NIN];
    int o = blockIdx.x, i = threadIdx.x;
    s[i] = Bias[o * NIN + i];
    __syncthreads();
    for (int st = NIN / 2; st > 0; st >>= 1) {
        if (i < st) s[i] += s[i + st];
        __syncthreads();
    }
    if (i == 0) bsum[o] = s[0];
}

// ---------------------------------------------------------------------------
// Kernel 3: postacts[b,o,i] = W[o,i]*x[b,i] + Bias[o,i]
// Bandwidth kernel: 512 MiB of nontemporal B128 stores; x/W/Bias live in L2.
// ---------------------------------------------------------------------------
__global__ void postacts_kernel(const float* __restrict__ x,
                                const float* __restrict__ W,
                                const float* __restrict__ Bias,
                                float* __restrict__ post) {
    long long t = (long long)blockIdx.x * blockDim.x + threadIdx.x;
    int i4 = (int)(t & 63);          // NIN/4 = 64 float4s per row
    int o  = (int)((t >> 6) & 255);
    int b  = (int)(t >> 14);

    f4 xv = ((const f4*)(x + (long long)b * NIN))[i4];
    f4 wv = ((const f4*)(W + (long long)o * NIN))[i4];
    f4 bv = ((const f4*)(Bias + (long long)o * NIN))[i4];

    f4 r;
    r.x = fmaf(wv.x, xv.x, bv.x);
    r.y = fmaf(wv.y, xv.y, bv.y);
    r.z = fmaf(wv.z, xv.z, bv.z);
    r.w = fmaf(wv.w, xv.w, bv.w);

    f4* dst = (f4*)(post + ((long long)b * NOUT + o) * NIN) + i4;
    __builtin_nontemporal_store(r, dst);
}

// ---------------------------------------------------------------------------
// Kernel 4: y = x @ W^T + bsum via V_WMMA_F32_16X16X4_F32.
// One wave32 per 16x16 output tile; K=256 consumed 4 at a time.
// A 16x4 f32: 2 VGPRs (lane m=lane&15, Koff=2*(lane>>4); A[0]=K+Koff, A[1]=K+Koff+1)
// B 4x16 f32: 2 VGPRs (lane n=lane&15, same K split) -> B[k][n] = W[o0+n][k]
// C/D 16x16 f32: 8 VGPRs (row = r + 8*(lane>>4), col = lane&15)
// ---------------------------------------------------------------------------
__global__ void gemm_wmma_kernel(const float* __restrict__ x,
                                 const float* __restrict__ W,
                                 const float* __restrict__ bsum,
                                 float* __restrict__ y) {
    int lane = threadIdx.x & 31;
    int wave = threadIdx.x >> 5;                 // 8 waves per block
    int tile = blockIdx.x * 8 + wave;            // 2048 tiles total
    int ob = tile & 15;                          // NOUT/16 = 16
    int bb = tile >> 4;                          // NB/16 = 128
    int b0 = bb * 16;
    int o0 = ob * 16;

    int m    = lane & 15;
    int koff = (lane >> 4) << 1;                 // 0 or 2

    const float* xrow = x + (long long)(b0 + m) * NIN;  // A row (m = batch)
    const float* wrow = W + (long long)(o0 + m) * NIN;  // B col (m = out index n)

    v8f acc = {};
#pragma unroll 8
    for (int k = 0; k < NIN; k += 4) {
        v2f a, bm;
        a.x  = xrow[k + koff];
        a.y  = xrow[k + koff + 1];
        bm.x = wrow[k + koff];
        bm.y = wrow[k + koff + 1];
        acc = __builtin_amdgcn_wmma_f32_16x16x4_f32(
            /*neg_a=*/false, a, /*neg_b=*/false, bm,
            /*c_mod=*/(short)0, acc, /*reuse_a=*/false, /*reuse_b=*/false);
    }

    float bs = bsum[o0 + m];                     // col index == lane&15
    int rbase = b0 + ((lane >> 4) << 3);         // +8 for upper half-wave
#pragma unroll
    for (int r = 0; r < 8; ++r) {
        y[(long long)(rbase + r) * NOUT + o0 + m] = acc[r] + bs;
    }
}

// ---------------------------------------------------------------------------
extern "C" void kernel_launch(void* const* d_in, const int* in_sizes, int n_in,
                              void* d_out, int out_size, void* d_ws, size_t ws_size,
                              hipStream_t stream) {
    const float* x      = (const float*)d_in[0];   // [2048,256]
    const float* affine = (const float*)d_in[1];   // [256,256,4]
    const float* mask   = (const float*)d_in[2];   // [256,256]

    float* y    = (float*)d_out;                       // [2048,256]
    float* post = (float*)d_out + (long long)NB * NOUT; // [2048,256,256]

    float* W    = (float*)d_ws;                    // 65536 floats
    float* Bias = W + NOUT * NIN;                  // 65536 floats
    float* bsum = Bias + NOUT * NIN;               // 256 floats

    // 1) fold affine+mask
    prep_kernel<<<(NOUT * NIN) / 256, 256, 0, stream>>>(affine, mask, W, Bias);

    // 2) bias row sums
    biassum_kernel<<<NOUT, NIN, 0, stream>>>(Bias, bsum);

    // 3) postacts (dominant: 512 MiB NT stores)
    long long nvec = (long long)NB * NOUT * (NIN / 4);  // 33,554,432 float4
    postacts_kernel<<<(int)(nvec / 256), 256, 0, stream>>>(x, W, Bias, post);

    // 4) y GEMM via f32 WMMA (2048 tiles, 8 waves/block)
    gemm_wmma_kernel<<<(NB / 16) * (NOUT / 16) / 8, 256, 0, stream>>>(x, W, bsum, y);
}